// MyModule_4879082848708
// MI455X (gfx1250) — compile-verified
//
#include <hip/hip_runtime.h>

// NMS (torchvision-style) for N=8192 boxes on gfx1250.
// Pipeline: rank-sort -> gather sorted boxes -> 16x16-tiled pairwise
// suppression mask (TDM tensor_load_to_lds stages box tiles per block,
// WMMA f32 16x16x4 computes area_i+area_j on the matrix pipe, VALU does the
// intersection, wave32 ballot packs bits branch-free via an LDS dump slot)
// -> serial greedy scan in one block with an in-LDS removed-bitmap and
// global_prefetch of the next mask row -> compacted output.
// Outputs (as float): segments 0..3 = plain NMS keep list (padded -1),
// segment 4 = class-offset (batched) NMS keep list.

#define NMS_THR 0.5f

typedef __attribute__((ext_vector_type(2))) float v2f;
typedef __attribute__((ext_vector_type(8))) float v8f;
typedef __attribute__((ext_vector_type(4))) unsigned int v4u;
typedef __attribute__((ext_vector_type(8))) int v8i;
typedef __attribute__((ext_vector_type(4))) int v4i;

// ---------------------------------------------------------------------------
// TDM: 1-D contiguous copy of n8 8-byte elements global -> LDS.
// D# per CDNA5 ISA ch.10.7/8: group0 = {count=1, lds_addr, global_addr(57b),
// type=2}; group1 = {wg_mask=0, data_size=3(8B), tensor_dim0=n8,
// tensor_dim1=1, tile_dim0=n8, tile_dim1/2=0(unused), stride0=n8};
// trailing groups zero (<=2D descriptor). Issued once per executing wave
// (EXEC ignored), tracked on the issuing wave's TENSORcnt.
// This toolchain's builtin is the 6-arg form:
//   (uint32x4, int32x8, int32x4, int32x4, int32x8, i32 cpol)
// ---------------------------------------------------------------------------
__device__ __forceinline__ void tdm_copy_1d(unsigned lds_addr,
                                            unsigned long long gaddr,
                                            unsigned n8) {
  v4u g0;
  v8i g1;
  v4i g2 = {0, 0, 0, 0}, g3 = {0, 0, 0, 0};
  v8i g4 = {0, 0, 0, 0, 0, 0, 0, 0};
  g0[0] = 1u;                                   // count=1, user descriptor
  g0[1] = lds_addr;                             // LDS byte address
  g0[2] = (unsigned)gaddr;                      // global_addr[31:0]
  g0[3] = (unsigned)((gaddr >> 32) & 0x01FFFFFFu) | (2u << 30);  // [56:32]|type=2
  g1[0] = (int)(3u << 16);                      // data_size=3 (8 bytes)
  g1[1] = (int)((n8 & 0xFFFFu) << 16);          // tensor_dim0[15:0]
  g1[2] = (int)((n8 >> 16) | (1u << 16));       // tensor_dim0[31:16] | tensor_dim1=1
  g1[3] = (int)(n8 << 16);                      // tensor_dim1 hi | tile_dim0=n8
  g1[4] = 0;                                    // tile_dim1=0, tile_dim2=0
  g1[5] = (int)n8;                              // tensor_dim0_stride[31:0]
  g1[6] = 0;
  g1[7] = 0;
  __builtin_amdgcn_tensor_load_to_lds(g0, g1, g2, g3, g4, 0);
}

// order[rank(i)] = i; rank counts (score>s_i) or (==, smaller idx):
// exactly jnp.argsort(-scores) with stable tie-break.
__global__ void rank_order_kernel(const float* __restrict__ scores,
                                  int* __restrict__ order, int n) {
  __shared__ float tile[256];
  int i = blockIdx.x * 256 + threadIdx.x;
  float si = scores[i];
  int rank = 0;
  for (int base = 0; base < n; base += 256) {
    tile[threadIdx.x] = scores[base + threadIdx.x];
    __syncthreads();
#pragma unroll 8
    for (int t = 0; t < 256; ++t) {
      float sj = tile[t];
      int j = base + t;
      rank += (int)((sj > si) || ((sj == si) && (j < i)));
    }
    __syncthreads();
  }
  order[rank] = i;
}

__global__ void init_maxc_kernel(unsigned* maxc) {
  if (threadIdx.x == 0 && blockIdx.x == 0) *maxc = 0u;
}

// max over all (non-negative) box coords; uint-ordered atomicMax is exact
// for non-negative floats.
__global__ void reduce_max_kernel(const float* __restrict__ x, int n,
                                  unsigned* __restrict__ maxc) {
  __shared__ float s[256];
  float m = 0.0f;
  for (int i = blockIdx.x * 256 + threadIdx.x; i < n; i += gridDim.x * 256)
    m = fmaxf(m, x[i]);
  s[threadIdx.x] = m;
  __syncthreads();
  for (int off = 128; off > 0; off >>= 1) {
    if (threadIdx.x < off) s[threadIdx.x] = fmaxf(s[threadIdx.x], s[threadIdx.x + off]);
    __syncthreads();
  }
  if (threadIdx.x == 0) atomicMax(maxc, __float_as_uint(s[0]));
}

__global__ void gather_kernel(const float4* __restrict__ boxes,
                              const int* __restrict__ order,
                              const int* __restrict__ idxs,
                              const unsigned* __restrict__ maxc,
                              float4* __restrict__ sboxes, int n, int batched) {
  int k = blockIdx.x * 256 + threadIdx.x;
  int src = order[k];
  float4 b = boxes[src];
  if (batched) {
    float mc = __uint_as_float(*maxc);
    float off = (float)idxs[src] * (mc + 1.0f);
    b.x += off; b.y += off; b.z += off; b.w += off;
  }
  sboxes[k] = b;
}

// One wave per 16x16 tile (8 waves/block); block covers 128 rows x 16 cols.
// TDM stages the block's 128 row boxes + 16 col boxes into LDS once.
// WMMA C-layout: lane L, vgpr r -> pair (row = r + 8*(L>=16), col = L&15).
// A = [area_row, 1, 0, 0] (16x4), B = [1; area_col; 0; 0] (4x16)
//   -> C[m][n] = area_row[m] + area_col[n].
// Suppression: inter*(1+T) > T*(a_i+a_j)  <=>  IoU > T  (union > 0 always).
__global__ void __launch_bounds__(256)
pair_mask_kernel(const float4* __restrict__ sboxes,
                 unsigned short* __restrict__ mask, int n) {
  const int WORDS = n >> 4;  // u16 words per mask row
  __shared__ float4 sBoxR[128];
  __shared__ float4 sBoxC[16];
  __shared__ float sAR[128];
  __shared__ float sAC[16];
  __shared__ unsigned short sW[8][18];  // [16] = dump slot for inactive lanes

  const int tid = threadIdx.x;
  const int lane = tid & 31;
  const int w = tid >> 5;
  const int ct = blockIdx.x;          // column tile
  const int rt = blockIdx.y * 8 + w;  // row tile (per wave)
  const int r0 = rt << 4;
  const int c0 = ct << 4;
  const int rb0 = blockIdx.y * 128;   // first row staged by this block

  // --- TDM staging: wave0 -> row boxes (256 qwords), wave1 -> col boxes ---
  if (w == 0) {
    tdm_copy_1d((unsigned)(uintptr_t)(void*)sBoxR,
                (unsigned long long)(uintptr_t)(sboxes + rb0), 256u);
  } else if (w == 1) {
    tdm_copy_1d((unsigned)(uintptr_t)(void*)sBoxC,
                (unsigned long long)(uintptr_t)(sboxes + c0), 32u);
  }
  if (w < 2) __builtin_amdgcn_s_wait_tensorcnt(0);
  __syncthreads();

  // Areas (thread-parallel over the staged tiles).
  if (tid < 128) {
    float4 b = sBoxR[tid];
    sAR[tid] = (b.z - b.x) * (b.w - b.y);
  }
  if (tid < 16) {
    float4 b = sBoxC[tid];
    sAC[tid] = (b.z - b.x) * (b.w - b.y);
  }
  __syncthreads();

  // Strictly-lower tiles have no (col > row) pairs: write zeros, exit.
  // ct/rt are wave-uniform and no barriers remain below -> safe wave exit.
  if (ct < rt) {
    if (lane < 16) mask[(size_t)(r0 + lane) * WORDS + ct] = 0;
    return;
  }

  const int rloc = w << 4;  // this wave's rows within the block staging

  // A 16x4 f32: lanes 0-15 hold M=lane, K=0 (v.x) / K=1 (v.y);
  // lanes 16-31 hold K=2/3 (all zero here). B 4x16 analogous.
  v2f a, bb;
  int m = lane & 15;
  if (lane < 16) {
    a.x = sAR[rloc + m]; a.y = 1.0f;
    bb.x = 1.0f;         bb.y = sAC[m];
  } else {
    a.x = 0.0f; a.y = 0.0f;
    bb.x = 0.0f; bb.y = 0.0f;
  }
  v8f c = {};
  // (neg_a, A, neg_b, B, c_mod, C, reuse_a, reuse_b)
  c = __builtin_amdgcn_wmma_f32_16x16x4_f32(false, a, false, bb,
                                            (short)0, c, false, false);

  const int cl = lane & 15;
  const float cx1 = sBoxC[cl].x, cy1 = sBoxC[cl].y;
  const float cx2 = sBoxC[cl].z, cy2 = sBoxC[cl].w;
  const int half8 = (lane >> 4) << 3;
#pragma unroll
  for (int r = 0; r < 8; ++r) {
    int rl = r + half8;
    float4 rb = sBoxR[rloc + rl];
    float iw = fmaxf(fminf(rb.z, cx2) - fmaxf(rb.x, cx1), 0.0f);
    float ih = fmaxf(fminf(rb.w, cy2) - fmaxf(rb.y, cy1), 0.0f);
    float inter = iw * ih;
    bool sup = (inter * (1.0f + NMS_THR) > NMS_THR * c[r]) &&
               ((c0 + cl) > (r0 + rl));
    unsigned bal = (unsigned)__ballot((int)sup);  // wave32: 32 valid bits
    // Branch-free staging: lane0 -> word for row r0+r, lane16 -> row r0+r+8,
    // all other lanes hit the dump slot (raced garbage, never read).
    int dst = (lane == 0) ? r : ((lane == 16) ? (r + 8) : 16);
    sW[w][dst] = (unsigned short)((lane < 16) ? (bal & 0xFFFFu) : (bal >> 16));
  }
  asm volatile("s_wait_dscnt 0" ::: "memory");
  // One store instruction: lanes 0-15 each write their row's 16-bit word.
  if (lane < 16) mask[(size_t)(r0 + lane) * WORDS + ct] = sW[w][lane];
}

// Serial greedy scan: one block, removed-bitmap (n/32 <= 256 words) in LDS.
// Prefetches the next mask row (global_prefetch) while OR-ing the current
// one. Writes compacted keep indices (as float) into nseg consecutive output
// segments starting at seg0, padded with -1.
__global__ void nms_scan_kernel(const unsigned short* __restrict__ mask,
                                const int* __restrict__ order,
                                float* __restrict__ out,
                                int n, int seg0, int nseg) {
  __shared__ unsigned rem[256];
  __shared__ int kc;
  const int t = threadIdx.x;
  rem[t] = 0u;
  if (t == 0) kc = 0;
  __syncthreads();
  const int words16 = n >> 4;
  for (int i = 0; i < n; ++i) {
    if (i + 1 < n)  // speculative: L2-resident row, hide the hop
      __builtin_prefetch(mask + (size_t)(i + 1) * words16 + 2 * t, 0, 0);
    bool alive = ((rem[i >> 5] >> (i & 31)) & 1u) == 0u;  // block-uniform
    if (alive) {
      if (t == 0) {
        float v = (float)order[i];
        for (int s = 0; s < nseg; ++s) out[(size_t)(seg0 + s) * n + kc] = v;
        kc++;
      }
      const unsigned short* row = mask + (size_t)i * words16;
      rem[t] |= ((unsigned)row[2 * t]) | (((unsigned)row[2 * t + 1]) << 16);
    }
    __syncthreads();
  }
  int K = kc;
  for (int p = K + t; p < n; p += 256)
    for (int s = 0; s < nseg; ++s) out[(size_t)(seg0 + s) * n + p] = -1.0f;
}

extern "C" void kernel_launch(void* const* d_in, const int* in_sizes, int n_in,
                              void* d_out, int out_size, void* d_ws, size_t ws_size,
                              hipStream_t stream) {
  const float*  boxesf = (const float*)d_in[0];
  const float4* boxes4 = (const float4*)d_in[0];
  const float*  scores = (const float*)d_in[1];
  const int*    idxs   = (const int*)d_in[2];
  float* out = (float*)d_out;
  const int n = in_sizes[1];  // 8192

  // Workspace: order[n] | sboxes[n]*16 | maxc(+pad) | mask u16[n][n/16]
  char* ws = (char*)d_ws;
  int*            order  = (int*)ws;
  float4*         sboxes = (float4*)(ws + (size_t)n * 4);
  unsigned*       maxc   = (unsigned*)(ws + (size_t)n * 4 + (size_t)n * 16);
  unsigned short* mask   = (unsigned short*)(ws + (size_t)n * 4 + (size_t)n * 16 + 256);

  dim3 tgrid(n / 16, n / 128);  // (col tiles, row-tile groups of 8 waves)

  rank_order_kernel<<<n / 256, 256, 0, stream>>>(scores, order, n);
  init_maxc_kernel<<<1, 32, 0, stream>>>(maxc);
  reduce_max_kernel<<<64, 256, 0, stream>>>(boxesf, 4 * n, maxc);

  // Plain NMS -> segments 0..3
  gather_kernel<<<n / 256, 256, 0, stream>>>(boxes4, order, idxs, maxc, sboxes, n, 0);
  pair_mask_kernel<<<tgrid, 256, 0, stream>>>(sboxes, mask, n);
  nms_scan_kernel<<<1, 256, 0, stream>>>(mask, order, out, n, 0, 4);

  // Batched (class-offset) NMS -> segment 4
  gather_kernel<<<n / 256, 256, 0, stream>>>(boxes4, order, idxs, maxc, sboxes, n, 1);
  pair_mask_kernel<<<tgrid, 256, 0, stream>>>(sboxes, mask, n);
  nms_scan_kernel<<<1, 256, 0, stream>>>(mask, order, out, n, 4, 1);
}